// Imputer_34016140985018
// MI455X (gfx1250) — compile-verified
//
#include <hip/hip_runtime.h>

// Imputer: x[B=64, D=2, N=2048, S=128] f32. For each (b,n): mean of valid
// entries of the first dim-row containing -inf fills ALL -inf slots at (b,n).
// Pure streaming kernel: 268 MB total traffic -> ~11.5 us at 23.3 TB/s.
//
// One full wave32 per (b,n): lane L holds elements [4L,4L+4) of both dim-rows.
// Counts -> SALU (ballot+popcount, exact integers, wave-uniform).
// Sum    -> matrix pipe: 5 chained v_wmma_f32_16x16x4_f32, no LDS, no shuffles.

typedef float v2f __attribute__((ext_vector_type(2)));
typedef float v4f __attribute__((ext_vector_type(4)));
typedef float v8f __attribute__((ext_vector_type(8)));

// 32-lane f32 sum done entirely on the WMMA pipe (exact f32 adds, RNE).
//
// A (16x4) layout: lane m (<16) holds A[m][0]=a.x, A[m][1]=a.y;
//                  lane m+16    holds A[m][2]=a.x, A[m][3]=a.y.
// Stage 1: a=(p,0), B=all-ones (layout-invariant) =>
//          D1[m][n] = p_m + p_{m+16}  (uniform across columns n).
// Stage 2: feed D1 back as A in natural VGPR pairs: with a=(d1[2k],d1[2k+1]),
//          lane c<16 supplies D1[2k][c],D1[2k+1][c]; lane c+16 supplies
//          D1[2k+8][c],D1[2k+9][c]. Four accumulating WMMAs therefore add
//          rows {0,1,8,9},{2,3,10,11},{4,5,12,13},{6,7,14,15} = all 16 rows:
//          D2[m][n] = sum_j D1[j][.] = sum over all 32 p, in every lane.
__device__ __forceinline__ float wave_sum32_wmma(float p) {
    v2f a;    a.x    = p;    a.y    = 0.0f;
    v2f ones; ones.x = 1.0f; ones.y = 1.0f;
    v8f z = {};
    v8f d1 = __builtin_amdgcn_wmma_f32_16x16x4_f32(
        false, a, false, ones, (short)0, z, false, false);
    v8f d2 = __builtin_amdgcn_wmma_f32_16x16x4_f32(
        false, __builtin_shufflevector(d1, d1, 0, 1), false, ones, (short)0, z,  false, false);
    d2 = __builtin_amdgcn_wmma_f32_16x16x4_f32(
        false, __builtin_shufflevector(d1, d1, 2, 3), false, ones, (short)0, d2, false, false);
    d2 = __builtin_amdgcn_wmma_f32_16x16x4_f32(
        false, __builtin_shufflevector(d1, d1, 4, 5), false, ones, (short)0, d2, false, false);
    d2 = __builtin_amdgcn_wmma_f32_16x16x4_f32(
        false, __builtin_shufflevector(d1, d1, 6, 7), false, ones, (short)0, d2, false, false);
    return d2[0];
}

__global__ __launch_bounds__(256)
void Imputer_34016140985018_kernel(const float* __restrict__ x,
                                   float* __restrict__ y) {
    // All waves fully populated (EXEC all-1s, required for WMMA).
    const unsigned tid  = blockIdx.x * 256u + threadIdx.x;
    const unsigned wave = tid >> 5;
    const unsigned lane = tid & 31u;
    const unsigned b    = wave >> 11;      // N = 2048
    const unsigned n    = wave & 2047u;

    // Element offsets (2^25 elements total: 32-bit indexing).
    const unsigned row0 = (b << 19) + (n << 7);  // ((b*2 + 0)*2048 + n) * 128
    const unsigned row1 = row0 + (1u << 18);     // + 2048*128 (dim stride)

    // 128-bit non-temporal streaming loads (touch-once data, skip L2 residency).
    v4f v0 = __builtin_nontemporal_load((const v4f*)(x + row0) + lane);
    v4f v1 = __builtin_nontemporal_load((const v4f*)(x + row1) + lane);

    const float NEG_INF = -__builtin_inff();

    // Valid counts on the SALU (ballot + popcount: exact, wave-uniform),
    // masked partial sums on the VALU.
    float    s0 = 0.0f, s1 = 0.0f;
    unsigned c0 = 0u,   c1 = 0u;
#pragma unroll
    for (int i = 0; i < 4; ++i) {
        const bool k0 = (v0[i] != NEG_INF);
        c0 += __builtin_popcount(__builtin_amdgcn_ballot_w32(k0));
        s0 += k0 ? v0[i] : 0.0f;
        const bool k1 = (v1[i] != NEG_INF);
        c1 += __builtin_popcount(__builtin_amdgcn_ballot_w32(k1));
        s1 += k1 ? v1[i] : 0.0f;
    }

    // d* = first dim with any missing (uniform scalar decision), so only ONE
    // float reduction is needed: reduce the selected row's partials.
    const bool     use0 = (c0 < 128u);       // row 0 has at least one -inf
    const float    psel = use0 ? s0 : s1;
    const unsigned csel = use0 ? c0 : c1;

    const float total = wave_sum32_wmma(psel);
    const float mean  = total / (float)(csel ? csel : 1u); // all--inf row -> 0

    // Fill every -inf slot (both dims) with the single (b,n) mean.
#pragma unroll
    for (int i = 0; i < 4; ++i) {
        v0[i] = (v0[i] != NEG_INF) ? v0[i] : mean;
        v1[i] = (v1[i] != NEG_INF) ? v1[i] : mean;
    }

    // 128-bit non-temporal streaming stores (output is write-once).
    __builtin_nontemporal_store(v0, (v4f*)(y + row0) + lane);
    __builtin_nontemporal_store(v1, (v4f*)(y + row1) + lane);
}

extern "C" void kernel_launch(void* const* d_in, const int* in_sizes, int n_in,
                              void* d_out, int out_size, void* d_ws, size_t ws_size,
                              hipStream_t stream) {
    (void)in_sizes; (void)n_in; (void)out_size; (void)d_ws; (void)ws_size;
    const float* x = (const float*)d_in[0];
    float*       y = (float*)d_out;

    // B*N = 64*2048 = 131072 (b,n) pairs; one wave32 each; 8 waves per block.
    const unsigned pairs  = 64u * 2048u;
    const unsigned blocks = pairs / 8u;   // 16384 blocks of 256 threads
    Imputer_34016140985018_kernel<<<dim3(blocks), dim3(256), 0, stream>>>(x, y);
}